// Slide_37838661878516
// MI455X (gfx1250) — compile-verified
//
#include <hip/hip_runtime.h>
#include <hip/hip_bf16.h>
#include <stdint.h>

#define NCOLS 8192          // N
#define BROWS 256           // B
#define TPB   256           // threads per block (8 wave32 waves)
#define SEG   2048          // columns per block in the main kernel
#define SEGS_PER_ROW (NCOLS / SEG)          // 4  -> grid = 4 x 256 = 1024 blocks
#define VEC_SEG (SEG / 4 / TPB)             // 2 float4 chunks per thread
#define VEC_ROW (NCOLS / 4 / TPB)           // 8 float4 chunks per thread (fallback)

// ---------------------------------------------------------------------------
// Pass 1: fold bias powers into the three weight diagonals.
//   A[r] = coef_sub[r]  * W[r, (r-1)%N]
//   D[r] = coef_diag[r] * W[r, r]
//   S[r] = coef_sup[r]  * W[r, (r+1)%N]
// Wrap columns r=0 and r=N-1 get the permuted (1+bias)^k powers.
// ---------------------------------------------------------------------------
__global__ __launch_bounds__(TPB) void slide_coeff_kernel(
    const float* __restrict__ W, const float* __restrict__ bias,
    float* __restrict__ A, float* __restrict__ D, float* __restrict__ S)
{
    const int n = NCOLS;
    const int r = blockIdx.x * blockDim.x + threadIdx.x;
    if (r >= n) return;

    const size_t row = (size_t)r * (size_t)n;
    const float w_sub  = W[row + (unsigned)(r + n - 1) % n];
    const float w_diag = W[row + r];
    const float w_sup  = W[row + (unsigned)(r + 1) % n];

    const float c1 = 1.0f + bias[r];
    const float c2 = c1 * c1;
    const float c3 = c2 * c1;

    float cs = c3, cd = c2, cp = c1;                 // interior columns
    if (r == 0)     { cs = c1; cd = c3; cp = c2; }   // wrap column 0
    if (r == n - 1) { cs = c2; cd = c1; cp = c3; }   // wrap column N-1

    A[r] = cs * w_sub;
    D[r] = cd * w_diag;
    S[r] = cp * w_sup;
}

// ---------------------------------------------------------------------------
// Async global->LDS staging (CDNA5 async data mover, ASYNCcnt-tracked).
// NVEC b128 copies per thread; caller must follow with the wait+barrier.
// ---------------------------------------------------------------------------
template <int NVEC>
__device__ __forceinline__ void stage_async(const float* __restrict__ g,
                                            float* __restrict__ s)
{
    const uint32_t lds_base = (uint32_t)(uintptr_t)(&s[0]); // low 32 bits == LDS offset
    const int t = threadIdx.x;
#pragma unroll
    for (int i = 0; i < NVEC; ++i) {
        const int e = (t + i * TPB) * 4;                    // float element index
        const uint32_t laddr = lds_base + (uint32_t)e * 4u;
        const unsigned long long gaddr = (unsigned long long)(uintptr_t)(g + e);
        asm volatile("global_load_async_to_lds_b128 %0, %1, off"
                     :: "v"(laddr), "v"(gaddr)
                     : "memory");
    }
}

__device__ __forceinline__ void wait_async_and_barrier()
{
    asm volatile("s_wait_asynccnt 0x0" ::: "memory");
    __syncthreads();
}

// ---------------------------------------------------------------------------
// Main kernel: grid (N/SEG, B) = (4, 256). Each block stages a 2048-column
// segment (8 KB) of one row into LDS asynchronously, plus two halo elements
// (global index mod N, so the circular wrap is exact). Compute is coalesced
// float4 FMAs.
// LDS layout: srow[0..SEG-1] = segment, srow[SEG] = x[base+SEG mod N] (right
// halo), srow[SEG+1] = x[base-1 mod N] (left halo).
// ---------------------------------------------------------------------------
__global__ __launch_bounds__(TPB) void slide_main_kernel(
    const float* __restrict__ x,
    const float* __restrict__ A, const float* __restrict__ D,
    const float* __restrict__ S, float* __restrict__ out)
{
    __shared__ __align__(16) float srow[SEG + 2];   // ~8.2 KB of 320 KB/WGP

    const int row  = blockIdx.y;
    const int base = blockIdx.x * SEG;
    const float* xrow = x + (size_t)row * NCOLS;

    stage_async<VEC_SEG>(xrow + base, srow);
    if (threadIdx.x == 0) {
        srow[SEG]     = xrow[(unsigned)(base + SEG)       & (NCOLS - 1)]; // right halo
        srow[SEG + 1] = xrow[(unsigned)(base + NCOLS - 1) & (NCOLS - 1)]; // left halo
    }
    wait_async_and_barrier();

    const float4* __restrict__ A4 = (const float4*)A;
    const float4* __restrict__ D4 = (const float4*)D;
    const float4* __restrict__ S4 = (const float4*)S;
    float4* __restrict__ o4 = (float4*)(out + (size_t)row * NCOLS + base);

    const int t = threadIdx.x;
#pragma unroll
    for (int i = 0; i < VEC_SEG; ++i) {
        const int jl = t + i * TPB;      // local float4 chunk index, 0..511
        const int lc = jl * 4;           // local column of chunk start
        const int jg = (base >> 2) + jl; // global float4 chunk index

        const float4 xv = *(const float4*)&srow[lc];
        const float xm1 = srow[(lc == 0)         ? (SEG + 1) : (lc - 1)];
        const float xp4 = srow[(lc + 4 == SEG)   ? SEG       : (lc + 4)];

        const float4 a = A4[jg];
        const float4 d = D4[jg];
        const float4 s = S4[jg];

        float4 r;
        r.x = fmaf(xm1,  a.x, fmaf(xv.x, d.x, xv.y * s.x));
        r.y = fmaf(xv.x, a.y, fmaf(xv.y, d.y, xv.z * s.y));
        r.z = fmaf(xv.y, a.z, fmaf(xv.z, d.z, xv.w * s.z));
        r.w = fmaf(xv.z, a.w, fmaf(xv.w, d.w, xp4  * s.w));
        o4[jl] = r;
    }
}

// ---------------------------------------------------------------------------
// Fallback (only if d_ws can't hold the 96 KB coefficient arrays): one full
// row per block, coefficients recomputed from weight/bias per column.
// Correct but does redundant L2 gathers; never expected to run in practice.
// ---------------------------------------------------------------------------
__global__ __launch_bounds__(TPB) void slide_fused_kernel(
    const float* __restrict__ x, const float* __restrict__ W,
    const float* __restrict__ bias, float* __restrict__ out)
{
    __shared__ __align__(16) float srow[NCOLS];     // full row: wrap is mod-index
    const int n = NCOLS;
    const int row = blockIdx.x;
    const float* xr = x + (size_t)row * NCOLS;

    stage_async<VEC_ROW>(xr, srow);
    wait_async_and_barrier();

    float4* __restrict__ o4 = (float4*)(out + (size_t)row * NCOLS);
    const int t = threadIdx.x;
#pragma unroll
    for (int i = 0; i < VEC_ROW; ++i) {
        const int j  = t + i * TPB;
        const int c0 = j * 4;
        float res[4];
#pragma unroll
        for (int cc = 0; cc < 4; ++cc) {
            const int r = c0 + cc;
            const size_t wrow = (size_t)r * (size_t)n;
            const float w_sub  = W[wrow + (unsigned)(r + n - 1) % n];
            const float w_diag = W[wrow + r];
            const float w_sup  = W[wrow + (unsigned)(r + 1) % n];
            const float c1 = 1.0f + bias[r];
            const float c2 = c1 * c1;
            const float c3 = c2 * c1;
            float cs = c3, cd = c2, cp = c1;
            if (r == 0)     { cs = c1; cd = c3; cp = c2; }
            if (r == n - 1) { cs = c2; cd = c1; cp = c3; }
            const float xm = srow[(unsigned)(r + n - 1) & (n - 1)];
            const float xc = srow[r];
            const float xp = srow[(unsigned)(r + 1) & (n - 1)];
            res[cc] = fmaf(xm, cs * w_sub, fmaf(xc, cd * w_diag, xp * (cp * w_sup)));
        }
        float4 rv; rv.x = res[0]; rv.y = res[1]; rv.z = res[2]; rv.w = res[3];
        o4[j] = rv;
    }
}

// ---------------------------------------------------------------------------
extern "C" void kernel_launch(void* const* d_in, const int* in_sizes, int n_in,
                              void* d_out, int out_size, void* d_ws, size_t ws_size,
                              hipStream_t stream)
{
    (void)in_sizes; (void)n_in; (void)out_size;
    const float* x    = (const float*)d_in[0];   // (B, N) f32
    const float* W    = (const float*)d_in[1];   // (N, N) f32
    const float* bias = (const float*)d_in[2];   // (N,)   f32
    float* out = (float*)d_out;                  // (B, N) f32

    const size_t need = 3 * (size_t)NCOLS * sizeof(float);   // 96 KB
    if (ws_size >= need) {
        float* A = (float*)d_ws;
        float* D = A + NCOLS;
        float* S = D + NCOLS;
        slide_coeff_kernel<<<NCOLS / TPB, TPB, 0, stream>>>(W, bias, A, D, S);
        dim3 grid(SEGS_PER_ROW, BROWS);
        slide_main_kernel<<<grid, TPB, 0, stream>>>(x, A, D, S, out);
    } else {
        slide_fused_kernel<<<BROWS, TPB, 0, stream>>>(x, W, bias, out);
    }
}